// FSEmbedLoss_51522427682923
// MI455X (gfx1250) — compile-verified
//
#include <hip/hip_runtime.h>
#include <hip/hip_bf16.h>
#include <math.h>

typedef float v2f __attribute__((ext_vector_type(2)));
typedef float v8f __attribute__((ext_vector_type(8)));

#define D_DIM  256
#define HW     16384
#define CNUM   19
#define CPAD   32
#define TILE   32
#define NTILES 4096      // 8*16384/32 pixel tiles
#define NBLK   512
#define FPAD   35        // LDS row pad: 35 odd, 2*35^-1 mod 64 = 22 -> no 2-way conflicts

// ---------------------------------------------------------------------------
// Pass 1: one sweep over the 128MB input.
//   gsum[c][d]  += sum of feats of class c           (onehot^T @ F   via WMMA f32)
//   gvsum[c][d] += sum of feats/||feat|| of class c  (onehot^T @ F*rn via WMMA f32)
//   gcnt[c]     += pixel count per class
// ---------------------------------------------------------------------------
__global__ __launch_bounds__(256) void fse_pass1(
    const float* __restrict__ in, const int* __restrict__ tgt,
    float* __restrict__ gsum, float* __restrict__ gvsum, int* __restrict__ gcnt)
{
  __shared__ float F[D_DIM * FPAD];   // tile transposed: [d][pixel]
  __shared__ float nsq[8 * 32];       // per-wave partial squared norms
  __shared__ float rn[TILE];          // 1/||f_p|| per pixel of tile
  __shared__ int   lab[TILE];
  __shared__ int   cnt[CPAD];

  const int tid  = threadIdx.x;
  const int lane = tid & 31;
  const int wid  = tid >> 5;          // wave id 0..7
  const int sel  = lane >> 4;         // lane half (K0/K1 vs K2/K3 rows)
  const int m    = lane & 15;

  if (tid < CPAD) cnt[tid] = 0;

  v8f zero;
  #pragma unroll
  for (int r = 0; r < 8; ++r) zero[r] = 0.f;
  v8f accS[2][2], accV[2][2];         // [d-tile][class-Mtile]
  #pragma unroll
  for (int a = 0; a < 2; ++a)
    #pragma unroll
    for (int b = 0; b < 2; ++b) { accS[a][b] = zero; accV[a][b] = zero; }

  for (int tile = blockIdx.x; tile < NTILES; tile += gridDim.x) {
    const int g0  = tile * TILE;
    const int b   = g0 >> 14;
    const int hw0 = g0 & (HW - 1);

    __syncthreads();                               // protect LDS reuse
    if (tid < TILE) lab[tid] = tgt[b * HW + hw0 + tid];

    // Coalesced load of 32 pixels x 256 dims; transpose into LDS; fuse |f|^2.
    float nacc = 0.f;
    const long base = ((long)b * D_DIM) * HW + hw0 + lane;
    #pragma unroll 4
    for (int j = 0; j < 32; ++j) {
      const int d = j * 8 + wid;
      const float v = in[base + (long)d * HW];
      F[d * FPAD + lane] = v;
      nacc = fmaf(v, v, nacc);
    }
    nsq[wid * 32 + lane] = nacc;
    __syncthreads();

    if (tid < TILE) {
      float s = 0.f;
      #pragma unroll
      for (int w2 = 0; w2 < 8; ++w2) s += nsq[w2 * 32 + tid];
      const float nf = sqrtf(s);
      rn[tid] = nf > 0.f ? 1.f / nf : 0.f;         // zero vector -> cos contribution 0
      atomicAdd(&cnt[lab[tid]], 1);
    }
    __syncthreads();

    // WMMA accumulate: 8 chunks of K=4 pixels. Wave w owns d-tiles {w, w+8}.
    // A/B VGPR halves: V0 = {K0 | K2}, V1 = {K1 | K3} per ISA layout.
    for (int c = 0; c < 8; ++c) {
      const int p0 = c * 4 + 2 * sel;              // pixel for even K row of this half
      const int p1 = p0 + 1;                       // pixel for odd K row
      const int l0 = lab[p0], l1 = lab[p1];
      const float r0 = rn[p0], r1 = rn[p1];
      v2f A0, A1;                                  // one-hot slices (exact in f32)
      A0.x = (l0 == m)        ? 1.f : 0.f;
      A0.y = (l1 == m)        ? 1.f : 0.f;
      A1.x = (l0 == m + 16)   ? 1.f : 0.f;
      A1.y = (l1 == m + 16)   ? 1.f : 0.f;
      #pragma unroll
      for (int t2 = 0; t2 < 2; ++t2) {
        const int drow = (wid + 8 * t2) * 16 + m;
        const float b0 = F[drow * FPAD + p0];
        const float b1 = F[drow * FPAD + p1];
        v2f Bs; Bs.x = b0;      Bs.y = b1;
        v2f Bv; Bv.x = b0 * r0; Bv.y = b1 * r1;
        accS[t2][0] = __builtin_amdgcn_wmma_f32_16x16x4_f32(false, A0, false, Bs, (short)0, accS[t2][0], false, false);
        accS[t2][1] = __builtin_amdgcn_wmma_f32_16x16x4_f32(false, A1, false, Bs, (short)0, accS[t2][1], false, false);
        accV[t2][0] = __builtin_amdgcn_wmma_f32_16x16x4_f32(false, A0, false, Bv, (short)0, accV[t2][0], false, false);
        accV[t2][1] = __builtin_amdgcn_wmma_f32_16x16x4_f32(false, A1, false, Bv, (short)0, accV[t2][1], false, false);
      }
    }
  }
  __syncthreads();

  // Flush register accumulators. C/D layout: VGPR r -> M = r + 8*(lane>=16), N = lane&15.
  #pragma unroll
  for (int t2 = 0; t2 < 2; ++t2) {
    const int d = (wid + 8 * t2) * 16 + m;
    #pragma unroll
    for (int mt = 0; mt < 2; ++mt) {
      #pragma unroll
      for (int r = 0; r < 8; ++r) {
        const int cls = mt * 16 + sel * 8 + r;
        atomicAdd(&gsum [cls * D_DIM + d], accS[t2][mt][r]);
        atomicAdd(&gvsum[cls * D_DIM + d], accV[t2][mt][r]);
      }
    }
  }
  if (tid < CPAD) atomicAdd(&gcnt[tid], cnt[tid]);
}

// ---------------------------------------------------------------------------
// Finalize: centroids, sim_loss (factored form), 19x19 gram -> diff_loss.
// ---------------------------------------------------------------------------
__global__ __launch_bounds__(256) void fse_finalize(
    const float* __restrict__ gsum, const float* __restrict__ gvsum,
    const int* __restrict__ gcnt, float* __restrict__ out)
{
  __shared__ float cen[CNUM * D_DIM];
  __shared__ float ncn[CNUM];
  __shared__ float simv[CNUM];
  __shared__ float peri[CNUM];
  __shared__ float cntf[CNUM];
  __shared__ int   prs[CNUM];

  const int tid = threadIdx.x;
  if (tid < CNUM) {
    const int c = gcnt[tid];
    prs[tid]  = (c > 0);
    cntf[tid] = fmaxf((float)c, 1.f);
    peri[tid] = 0.f;
    simv[tid] = 0.f;
  }
  __syncthreads();
  for (int i = tid; i < CNUM * D_DIM; i += 256)
    cen[i] = gsum[i] / cntf[i >> 8];               // both row-strides are 256
  __syncthreads();

  if (tid < CNUM) {
    float s2 = 0.f, dv = 0.f;
    for (int d = 0; d < D_DIM; ++d) {
      const float c = cen[tid * D_DIM + d];
      s2 += c * c;
      dv += gvsum[tid * D_DIM + d] * c;
    }
    const float nc = sqrtf(s2);
    ncn[tid] = nc;
    const float sumcos = (nc > 0.f) ? dv / nc : 0.f;   // sum over pixels of cos
    simv[tid] = prs[tid] ? (cntf[tid] - sumcos) / cntf[tid] : 0.f;
  }
  __syncthreads();

  for (int t = tid; t < CNUM * CNUM; t += 256) {
    const int i = t / CNUM, j = t % CNUM;
    float dv = 0.f;
    for (int d = 0; d < D_DIM; ++d) dv += cen[i * D_DIM + d] * cen[j * D_DIM + d];
    const float S = dv / fmaxf(ncn[i] * ncn[j], 1e-8f);
    const float v = (i == j) ? (1.f - S) : fmaxf(S, 0.f);
    atomicAdd(&peri[i], v * (1.f / (float)CNUM));
  }
  __syncthreads();

  if (tid == 0) {
    float tot = 0.f;
    for (int i = 0; i < CNUM; ++i) tot += simv[i] + (prs[i] ? peri[i] : 0.f);
    out[0] = tot;
  }
}

extern "C" void kernel_launch(void* const* d_in, const int* in_sizes, int n_in,
                              void* d_out, int out_size, void* d_ws, size_t ws_size,
                              hipStream_t stream) {
  const float* in  = (const float*)d_in[0];
  const int*   tgt = (const int*)d_in[1];
  float* out   = (float*)d_out;
  float* gsum  = (float*)d_ws;                 // [32][256]
  float* gvsum = gsum + CPAD * D_DIM;          // [32][256]
  int*   gcnt  = (int*)(gvsum + CPAD * D_DIM); // [32]

  hipMemsetAsync(d_ws, 0,
                 (size_t)(2 * CPAD * D_DIM) * sizeof(float) + CPAD * sizeof(int),
                 stream);
  fse_pass1<<<NBLK, 256, 0, stream>>>(in, tgt, gsum, gvsum, gcnt);
  fse_finalize<<<1, 256, 0, stream>>>(gsum, gvsum, gcnt, out);
}